// MeinBlock_5102421148166
// MI455X (gfx1250) — compile-verified
//
#include <hip/hip_runtime.h>

typedef __attribute__((ext_vector_type(2))) float v2f;
typedef __attribute__((ext_vector_type(8))) float v8f;
typedef __attribute__((ext_vector_type(4))) unsigned int u32x4;
typedef __attribute__((ext_vector_type(8))) int i32x8;
typedef __attribute__((ext_vector_type(4))) int i32x4;

#define DCH 128          // channels (D == H == 128)
#define APITCH 132       // floats per A row in LDS (TDM pad: 4 dwords per 128)
#define WP2 144          // float2 per W k-pair row; 2*144 % 64 == 32 -> conflict-free

#if __has_builtin(__builtin_amdgcn_tensor_load_to_lds)
#define HAVE_TDM 1
#else
#define HAVE_TDM 0
#endif

// ---------------------------------------------------------------------------
// zero a small float region (column stats + zero bias row)
__global__ void zero_small_kernel(float* p, int cnt) {
    int i = blockIdx.x * blockDim.x + threadIdx.x;
    if (i < cnt) p[i] = 0.0f;
}

// ---------------------------------------------------------------------------
// column sums of prelu(x): colsum[c] += sum_r v, colsq[c] += sum_r v*v
__global__ __launch_bounds__(DCH) void stats_kernel(
    const float* __restrict__ x, const float* __restrict__ a1,
    float* __restrict__ colsum, float* __restrict__ colsq, int n) {
    const int c = threadIdx.x;                 // 128 threads = 128 columns
    const float al = a1[0];
    int rows_per = (n + gridDim.x - 1) / gridDim.x;
    int r0 = blockIdx.x * rows_per;
    int r1 = r0 + rows_per; if (r1 > n) r1 = n;
    float s = 0.0f, s2 = 0.0f;
    for (int r = r0; r < r1; ++r) {
        float v = x[(size_t)r * DCH + c];
        v = (v >= 0.0f) ? v : al * v;
        s += v; s2 += v * v;
    }
    atomicAdd(&colsum[c], s);
    atomicAdd(&colsq[c], s2);
}

// ---------------------------------------------------------------------------
// degree: init to 1 (self loop), count edges, then rsqrt in place
__global__ void deg_init_kernel(float* deg, int n) {
    int i = blockIdx.x * blockDim.x + threadIdx.x;
    if (i < n) deg[i] = 1.0f;
}
__global__ void deg_count_kernel(const int* __restrict__ dst, float* deg, int e) {
    int i = blockIdx.x * blockDim.x + threadIdx.x;
    if (i < e) atomicAdd(&deg[dst[i]], 1.0f);
}
__global__ void deg_rsqrt_kernel(float* deg, int n) {
    int i = blockIdx.x * blockDim.x + threadIdx.x;
    if (i < n) deg[i] = rsqrtf(deg[i]);       // deg[] now holds dinv
}

// ---------------------------------------------------------------------------
// fold batchnorm into W1:  W1p[k][j] = scale[k]*W1[k][j],  c1[j] = shift@W1
__global__ __launch_bounds__(DCH) void finalize_fold_kernel(
    const float* __restrict__ colsum, const float* __restrict__ colsq,
    const float* __restrict__ gamma, const float* __restrict__ beta,
    const float* __restrict__ W1, float* __restrict__ W1p,
    float* __restrict__ c1row, int n) {
    __shared__ float ss[DCH], sh[DCH];
    const int c = threadIdx.x;
    const float invn = 1.0f / (float)n;
    float mean = colsum[c] * invn;
    float var  = colsq[c] * invn - mean * mean;  // biased var, matches reference
    float sc = gamma[c] * rsqrtf(var + 1e-5f);
    float sf = beta[c] - mean * sc;
    ss[c] = sc; sh[c] = sf;
    __syncthreads();
    float acc = 0.0f;
    for (int k = 0; k < DCH; ++k) {
        float w = W1[k * DCH + c];
        W1p[k * DCH + c] = ss[k] * w;
        acc += sh[k] * w;
    }
    c1row[c] = acc;
}

// ---------------------------------------------------------------------------
// H = prelu(X, alpha) @ W + brow     (128x128 tile per block, fp32 WMMA)
// block = 256 threads = 8 waves; wave w computes rows [w*16, w*16+16)
// A-tile is DMA'd into LDS by the Tensor Data Mover (pitch 132 via TDM pad);
// W is staged k-pair interleaved so each B fragment is a single ds_load_b64.
__global__ __launch_bounds__(256) void gemm_prelu_wmma_kernel(
    const float* __restrict__ X, const float* __restrict__ Wm,
    const float* __restrict__ brow, const float* __restrict__ alpha,
    float* __restrict__ Hout, int n) {
    __shared__ float sA[DCH * APITCH];           // 67,584 B
    __shared__ float sW2[64 * WP2 * 2];          // 73,728 B (interleaved W)
    const int t = threadIdx.x;
    const float al = alpha[0];
    const int rowbase = blockIdx.x * DCH;

#if HAVE_TDM
    if (t < 32) {   // wave 0 issues the tensor DMA: global X tile -> sA
        unsigned lds_addr = (unsigned)(uintptr_t)(void*)sA;   // low32 = LDS offset
        unsigned long long ga =
            (unsigned long long)(uintptr_t)(X + (size_t)rowbase * DCH);
        unsigned dim1 = (unsigned)(n - rowbase); // rows in tensor; OOB rows -> 0
        // D# group0: count=1 | lds_addr | global_addr | type=2
        u32x4 g0 = { 1u, lds_addr, (unsigned)ga,
                     (unsigned)(ga >> 32) | 0x80000000u };
        // D# group1: data_size=4B, pad_enable, pad_interval=128dw, pad_amount=4dw,
        //            tensor_dim0=128, tensor_dim1=dim1, tile=128x128, stride0=128
        i32x8 g1 = { (int)((2u << 16) | (1u << 20) | (6u << 22) | (3u << 25)),
                     (int)(128u << 16),            // tensor_dim0[15:0]
                     (int)((dim1 & 0xffffu) << 16),// dim0[31:16]=0 | dim1[15:0]
                     (int)((dim1 >> 16) | (128u << 16)), // dim1[31:16] | tile_dim0
                     128,                          // tile_dim1 (tile_dim2=0)
                     128,                          // tensor_dim0_stride[31:0]
                     0, 0 };
        i32x4 gz  = { 0, 0, 0, 0 };
        i32x8 gz8 = { 0, 0, 0, 0, 0, 0, 0, 0 };
        // clang-23 / therock-10.0 form: 6 args (extra int32x8 group before cpol)
        __builtin_amdgcn_tensor_load_to_lds(g0, g1, gz, gz, gz8, 0);
    }
#else
    #pragma unroll 4
    for (int i = 0; i < 64; ++i) {               // fallback: manual A staging
        int idx = t + i * 256;
        int r = idx >> 7, c = idx & 127;
        int gr = rowbase + r; if (gr >= n) gr = n - 1;
        sA[r * APITCH + c] = X[(size_t)gr * DCH + c];
    }
#endif

    // stage W interleaved by k-pair: sW2[(kp*WP2 + j)*2 + (k&1)] = W[k][j]
    #pragma unroll 4
    for (int i = 0; i < 64; ++i) {
        int idx = t + i * 256;
        int k = idx >> 7, j = idx & 127;
        sW2[(((k >> 1) * WP2) + j) * 2 + (k & 1)] = Wm[idx];
    }

#if HAVE_TDM
    if (t < 32) __builtin_amdgcn_s_wait_tensorcnt(0);
#endif
    __syncthreads();

    const int wave  = t >> 5;
    const int lane  = t & 31;
    const int l15   = lane & 15;
    const int khalf = (lane >> 4) << 1;          // 0 for lanes 0-15, 2 for 16-31
    const int arow  = wave * 16 + l15;

    v8f acc[8];
    #pragma unroll
    for (int nt = 0; nt < 8; ++nt) {
        float bb = brow[nt * 16 + l15];          // bias depends on column only
        #pragma unroll
        for (int i = 0; i < 8; ++i) acc[nt][i] = bb;
    }

    for (int k0 = 0; k0 < 32; ++k0) {            // K = 128 in steps of 4
        const int kk = (k0 << 2) + khalf;
        v2f a = *(const v2f*)&sA[arow * APITCH + kk];   // A: M=arow, K=kk,kk+1
        a.x = (a.x >= 0.0f) ? a.x : al * a.x;           // fused PReLU
        a.y = (a.y >= 0.0f) ? a.y : al * a.y;
        const int kp = kk >> 1;                  // k-pair index
        #pragma unroll
        for (int nt = 0; nt < 8; ++nt) {         // single b64, even-aligned pair
            v2f b = *(const v2f*)&sW2[(kp * WP2 + nt * 16 + l15) * 2];
            acc[nt] = __builtin_amdgcn_wmma_f32_16x16x4_f32(
                false, a, false, b, (short)0, acc[nt], false, false);
        }
    }

    // store D: VGPR v -> row (v) for lanes 0-15, row (v+8) for lanes 16-31
    const int rH = (lane >> 4) << 3;
    for (int nt = 0; nt < 8; ++nt) {
        int col = nt * 16 + l15;
        #pragma unroll
        for (int v = 0; v < 8; ++v) {
            int gr = rowbase + wave * 16 + v + rH;
            if (gr < n) Hout[(size_t)gr * DCH + col] = acc[nt][v];
        }
    }
}

// ---------------------------------------------------------------------------
// Out[i][c] = H[i][c] * dinv[i]^2 + bias[c]   (self-loop term + conv bias)
__global__ void agg_init_kernel(const float* __restrict__ H,
                                const float* __restrict__ dinv,
                                const float* __restrict__ bias,
                                float* __restrict__ Out, int n) {
    int idx = blockIdx.x * blockDim.x + threadIdx.x;
    if (idx < n * DCH) {
        int i = idx >> 7, c = idx & 127;
        float di = dinv[i];
        Out[idx] = H[idx] * di * di + bias[c];
    }
}

// ---------------------------------------------------------------------------
// edge scatter: Out[dst] += H[src] * (dinv[src]*dinv[dst]); one wave per edge
__global__ __launch_bounds__(256) void scatter_edges_kernel(
    const float* __restrict__ H, const float* __restrict__ dinv,
    const int* __restrict__ src, const int* __restrict__ dst,
    float* __restrict__ Out, int e) {
    const int gtid = blockIdx.x * blockDim.x + threadIdx.x;
    const int wave = gtid >> 5;
    const int lane = threadIdx.x & 31;
    const int nw   = (gridDim.x * blockDim.x) >> 5;
    for (int ed = wave; ed < e; ed += nw) {
        int s = src[ed], d = dst[ed];
        if (ed + nw < e) {  // prefetch next source row (global_prefetch_b8)
            int ns = src[ed + nw];
            __builtin_prefetch(H + (size_t)ns * DCH + lane * 4, 0, 1);
        }
        float coef = dinv[s] * dinv[d];
        float4 v = ((const float4*)(H + (size_t)s * DCH))[lane];  // 512B row
        float* o = Out + (size_t)d * DCH + lane * 4;
        atomicAdd(o + 0, v.x * coef);
        atomicAdd(o + 1, v.y * coef);
        atomicAdd(o + 2, v.z * coef);
        atomicAdd(o + 3, v.w * coef);
    }
}

// ---------------------------------------------------------------------------
extern "C" void kernel_launch(void* const* d_in, const int* in_sizes, int n_in,
                              void* d_out, int out_size, void* d_ws, size_t ws_size,
                              hipStream_t stream) {
    const float* x     = (const float*)d_in[0];
    const int*   ei    = (const int*)d_in[1];
    const float* a1    = (const float*)d_in[2];
    const float* gamma = (const float*)d_in[3];
    const float* beta  = (const float*)d_in[4];
    const float* W1    = (const float*)d_in[5];
    const float* b1    = (const float*)d_in[6];
    const float* a2    = (const float*)d_in[7];
    const float* W2    = (const float*)d_in[8];
    const float* b2    = (const float*)d_in[9];
    const int n = in_sizes[0] / DCH;
    const int e = in_sizes[1] / 2;
    const int* src = ei;
    const int* dst = ei + e;

    // workspace layout (floats)
    float* ws     = (float*)d_ws;
    float* Hbuf   = ws;                          // N*128
    float* deg    = Hbuf + (size_t)n * DCH;      // N      (becomes dinv)
    float* colsum = deg + n;                     // 128
    float* colsq  = colsum + DCH;                // 128
    float* zrow   = colsq + DCH;                 // 128 (zero bias row)
    float* W1p    = zrow + DCH;                  // 128*128 folded W1
    float* c1row  = W1p + DCH * DCH;             // 128 folded bias row
    float* out    = (float*)d_out;

    zero_small_kernel<<<2, 256, 0, stream>>>(colsum, 3 * DCH);
    stats_kernel<<<512, DCH, 0, stream>>>(x, a1, colsum, colsq, n);
    deg_init_kernel<<<(n + 255) / 256, 256, 0, stream>>>(deg, n);
    deg_count_kernel<<<(e + 255) / 256, 256, 0, stream>>>(dst, deg, e);
    finalize_fold_kernel<<<1, DCH, 0, stream>>>(colsum, colsq, gamma, beta, W1,
                                                W1p, c1row, n);
    deg_rsqrt_kernel<<<(n + 255) / 256, 256, 0, stream>>>(deg, n);

    const int gemm_blocks = (n + 127) / 128;
    const int elem_blocks = (n * DCH + 255) / 256;

    // conv1: H1 = prelu(x,a1) @ W1' + c1  ; out = agg(H1) + b1
    gemm_prelu_wmma_kernel<<<gemm_blocks, 256, 0, stream>>>(x, W1p, c1row, a1, Hbuf, n);
    agg_init_kernel<<<elem_blocks, 256, 0, stream>>>(Hbuf, deg, b1, out, n);
    scatter_edges_kernel<<<2048, 256, 0, stream>>>(Hbuf, deg, src, dst, out, e);

    // conv2: H2 = prelu(out,a2) @ W2 ; out = agg(H2) + b2
    gemm_prelu_wmma_kernel<<<gemm_blocks, 256, 0, stream>>>(out, W2, zrow, a2, Hbuf, n);
    agg_init_kernel<<<elem_blocks, 256, 0, stream>>>(Hbuf, deg, b2, out, n);
    scatter_edges_kernel<<<2048, 256, 0, stream>>>(Hbuf, deg, src, dst, out, e);
}